// RANGE2BEV_35931696399119
// MI455X (gfx1250) — compile-verified
//
#include <hip/hip_runtime.h>

// ---------------------------------------------------------------------------
// Range-image -> BEV scatter (B=2, C=64, RH=64, RW=2048 -> out 2x64x5x400x352)
// Bandwidth-bound: ~360MB NT write (once, coalesced) + ~67MB L2-resident gather.
// No contraction in this op -> WMMA inapplicable; the CDNA5 paths used are
// 128-bit non-temporal global stores, global_prefetch_b8, wave32 ballot for a
// wave-uniform empty-cell fast path, and deterministic atomicMax election.
// ---------------------------------------------------------------------------

#define B_  2
#define C_  64
#define RH_ 64
#define RW_ 2048
#define NPTS (RH_ * RW_)        // 131072 points per batch
#define W_  400                 // (40 - -40)/0.2
#define H_  352                 // (70.4 - 0)/0.2
#define D_  5                   // (1 - -3)/0.8
#define DWH (D_ * W_ * H_)      // 704000 cells per batch
#define TOTAL_CELLS (B_ * DWH)  // 1408000

typedef float v4f __attribute__((ext_vector_type(4)));
typedef int   v4i __attribute__((ext_vector_type(4)));

// ---------------------------------------------------------------------------
// Pass 1: per-point winner election. Deterministic atomicMax of point-index
// into the target cell (max-index wins == a valid, replay-deterministic
// instance of JAX's unspecified duplicate-scatter order, mode='drop').
// ---------------------------------------------------------------------------
__global__ void __launch_bounds__(256)
bev_winner_kernel(const float* __restrict__ xyz, int* __restrict__ winner) {
    int t = blockIdx.x * blockDim.x + threadIdx.x;
    if (t >= B_ * NPTS) return;
    int b = t >> 17;                 // / NPTS (power of two)
    int n = t & (NPTS - 1);

    const float* base = xyz + (size_t)b * 3 * NPTS;
    float x = base[n];
    float y = base[(size_t)NPTS + n];
    float z = base[(size_t)2 * NPTS + n];

    // valid = (z >= -3) & (z < 1); invalid -> dropped
    if (!(z >= -3.0f && z < 1.0f)) return;
    int zbin = (int)floorf((z + 3.0f) / 0.8f);
    if (zbin < 0 || zbin >= D_) return;      // mode='drop' (fp edge at z->1 too)

    // x_img = trunc(-y/0.2) + 200  clip [0,399];  y_img = trunc(-x/0.2) + 352  clip [0,351]
    int x_img = (int)(-y / 0.2f) + 200;      // C cast truncates == astype(int32)
    int y_img = (int)(-x / 0.2f) + 352;
    x_img = min(max(x_img, 0), W_ - 1);
    y_img = min(max(y_img, 0), H_ - 1);
    int xf = (W_ - 1) - x_img;
    int yf = (H_ - 1) - y_img;

    int cell = b * DWH + (zbin * W_ + xf) * H_ + yf;
    atomicMax(&winner[cell], n);             // global_atomic_max_i32 (no return)
}

// ---------------------------------------------------------------------------
// Pass 2: output-stationary fill. One thread owns 4 adjacent H cells, loops
// over 64 channels, one 128-bit NON-TEMPORAL store per channel: the 360MB
// output is written exactly once, fully coalesced, without L2 retention
// (output >> 192MB L2, never re-read). Occupied-cell gathers are random in n
// but the 67MB feature tensor is L2-resident. Wave-uniform ballot fast path
// skips all gather work for fully-empty wave tiles (~80% of cells are empty).
// ---------------------------------------------------------------------------
__global__ void __launch_bounds__(256)
bev_fill_kernel(const float* __restrict__ feat,     // [B, C, NPTS]
                const int*   __restrict__ winner,   // [B * DWH]
                float*       __restrict__ out) {    // [B, C, DWH]
    int t = blockIdx.x * blockDim.x + threadIdx.x;
    if (t >= TOTAL_CELLS / 4) return;

    int cell0 = t * 4;                       // 16B aligned; H=352 % 4 == 0
    int b = (cell0 >= DWH) ? 1 : 0;          // B_ == 2: compare beats idiv
    int r = cell0 - b * DWH;                 // spatial offset inside batch slab

    v4i win = *(const v4i*)(winner + cell0);

    float* ob = out + (size_t)b * C_ * DWH + r;

    // all-four-negative <=> sign bit survives the AND
    bool occupied = ((win.x & win.y & win.z & win.w) >= 0) ||
                    (win.x >= 0) || (win.y >= 0) || (win.z >= 0) || (win.w >= 0);
    // (the AND test alone is exact: all<0 iff AND<0; keep it simple & exact:)
    occupied = !((win.x < 0) & (win.y < 0) & (win.z < 0) & (win.w < 0));

    if (__builtin_amdgcn_ballot_w32(occupied) == 0u) {
        // Wave-uniform empty path: pure NT zero-fill, no gathers, no selects.
        v4f zv = {0.0f, 0.0f, 0.0f, 0.0f};
        #pragma unroll 4
        for (int c = 0; c < C_; ++c)
            __builtin_nontemporal_store(zv, (v4f*)(ob + (size_t)c * DWH));
        return;
    }

    // Branchless gathers: clamp index to 0 (always-valid L2-hit), select after.
    const float* fb = feat + (size_t)b * C_ * NPTS;
    int ix = max(win.x, 0), iy = max(win.y, 0);
    int iz = max(win.z, 0), iw = max(win.w, 0);

    #pragma unroll 4
    for (int c = 0; c < C_; ++c) {
        const float* fc = fb + (size_t)c * NPTS;

        if (c + 1 < C_) {                    // hide the 512KB channel stride
            __builtin_prefetch(fc + NPTS + ix, 0, 0);
            __builtin_prefetch(fc + NPTS + iw, 0, 0);
        }

        v4f v;
        v.x = (win.x >= 0) ? fc[ix] : 0.0f;  // unconditional load + v_cndmask
        v.y = (win.y >= 0) ? fc[iy] : 0.0f;
        v.z = (win.z >= 0) ? fc[iz] : 0.0f;
        v.w = (win.w >= 0) ? fc[iw] : 0.0f;

        __builtin_nontemporal_store(v, (v4f*)(ob + (size_t)c * DWH));
    }
}

// ---------------------------------------------------------------------------
extern "C" void kernel_launch(void* const* d_in, const int* in_sizes, int n_in,
                              void* d_out, int out_size, void* d_ws, size_t ws_size,
                              hipStream_t stream) {
    const float* range_res = (const float*)d_in[0];   // [B, C, RH, RW] f32
    const float* xyz       = (const float*)d_in[1];   // [B, 3, RH, RW] f32
    (void)in_sizes; (void)n_in; (void)out_size; (void)ws_size;

    int* winner = (int*)d_ws;                         // B*DWH ints = 5.6 MB

    // winner[] = -1 everywhere (0xFF bytes); graph-capturable memset node.
    hipMemsetAsync(winner, 0xFF, (size_t)TOTAL_CELLS * sizeof(int), stream);

    {
        int threads = 256;
        int blocks  = (B_ * NPTS + threads - 1) / threads;        // 1024
        bev_winner_kernel<<<blocks, threads, 0, stream>>>(xyz, winner);
    }
    {
        int threads = 256;
        int blocks  = (TOTAL_CELLS / 4 + threads - 1) / threads;  // 1375
        bev_fill_kernel<<<blocks, threads, 0, stream>>>(range_res, winner,
                                                        (float*)d_out);
    }
}